// UniformLaplacian_8461085573740
// MI455X (gfx1250) — compile-verified
//
#include <hip/hip_runtime.h>

// ---- problem constants (match reference: B, NV, F, D = 4, 100000, 200000, 64) ----
constexpr int       kB          = 4;
constexpr int       kNV         = 100000;
constexpr int       kF          = 200000;
constexpr int       kD          = 64;          // floats per vertex feature row
constexpr int       kC4         = kD / 4;      // float4 chunks per row = 16
constexpr long long kFacesTotal = (long long)kB * kF;        // 800,000
constexpr long long kVertsTotal = (long long)kB * kNV;       // 400,000
constexpr long long kOutFloats  = kVertsTotal * kD;          // 25,600,000

constexpr int kBlock        = 256;
constexpr int kFacesPerBlk  = kBlock / kC4;     // 16 faces per block
constexpr int kIntsPerBlk   = kFacesPerBlk * 3; // 48 face indices per block

typedef float f32x4 __attribute__((ext_vector_type(4)));

// Typed address-space casts for the gfx1250 async global->LDS builtin
// (diagnostic showed param 0 is a non-const global-AS int*).
#define AS_GLOBAL_INT(p) ((__attribute__((address_space(1))) int*)(p))
#define AS_LDS_INT(p)    ((__attribute__((address_space(3))) int*)(p))

#if __has_builtin(__builtin_amdgcn_global_load_async_to_lds_b32)
#define UL_HAVE_ASYNC_LDS 1
#else
#define UL_HAVE_ASYNC_LDS 0
#endif

// -------------------------------------------------------------------------
// Kernel 1: zero the neighbor accumulator (aliased onto d_out) and degrees.
// Pure b128 streaming stores (lines stay dirty in L2 for the atomic phase).
// -------------------------------------------------------------------------
__global__ void ul_zero_kernel(float4* __restrict__ out4, int nOut4,
                               float4* __restrict__ deg4, int nDeg4) {
  const int stride = gridDim.x * blockDim.x;
  const float4 z = make_float4(0.f, 0.f, 0.f, 0.f);
  for (int i = blockIdx.x * blockDim.x + threadIdx.x; i < nOut4; i += stride)
    out4[i] = z;
  for (int i = blockIdx.x * blockDim.x + threadIdx.x; i < nDeg4; i += stride)
    deg4[i] = z;
}

// -------------------------------------------------------------------------
// Kernel 2: per-face accumulation.
// * Block stages its 48 face indices into LDS once (async global->LDS copy
//   on gfx1250; ASYNCcnt + s_wait_asynccnt + barrier).
// * 16 threads per face; thread c owns float4 chunk c: three 256B row
//   gathers (global_load_b128) and 12 hardware f32 atomics with pair sums
//   (global_atomic_add_f32 no-return -> STOREcnt path).
// -------------------------------------------------------------------------
__global__ void __launch_bounds__(kBlock)
ul_accum_kernel(const float* __restrict__ verts,
                const int*   __restrict__ faces,
                float*       __restrict__ nbr,   // == d_out
                float*       __restrict__ deg) {
  __shared__ int sf[kIntsPerBlk];

  const int t = threadIdx.x;
  const long long blockFaceBase = (long long)blockIdx.x * kFacesPerBlk;

  // ---- stage face indices for this block's 16 faces into LDS ----
  if (t < kIntsPerBlk) {
    const int* gsrc = faces + blockFaceBase * 3 + t;
#if UL_HAVE_ASYNC_LDS
    __builtin_amdgcn_global_load_async_to_lds_b32(AS_GLOBAL_INT(gsrc),
                                                  AS_LDS_INT(&sf[t]),
                                                  /*offset=*/0, /*cpol=*/0);
#else
    sf[t] = *gsrc;
#endif
    // Pull the face-index stream ahead into cache (global_prefetch_b8).
    __builtin_prefetch(gsrc + kIntsPerBlk * 32, 0, 1);
  }
#if UL_HAVE_ASYNC_LDS
  asm volatile("s_wait_asynccnt 0x0" ::: "memory");
#endif
  __syncthreads();

  // ---- per-(face, chunk) work ----
  const int fl   = (t >> 4);           // face within block: 0..15
  const int c    = t & 15;             // float4 chunk within row: 0..15
  const int face = (int)blockFaceBase + fl;
  const int b    = face / kF;
  const int base = b * kNV;            // block-diagonal offset into [B*NV]

  const int i0 = base + sf[fl * 3 + 0];
  const int i1 = base + sf[fl * 3 + 1];
  const int i2 = base + sf[fl * 3 + 2];

  const float4* __restrict__ v4 = (const float4*)verts;
  const float4 a0 = v4[(long long)i0 * kC4 + c];
  const float4 a1 = v4[(long long)i1 * kC4 + c];
  const float4 a2 = v4[(long long)i2 * kC4 + c];

  float* w0 = nbr + (long long)i0 * kD + c * 4;
  float* w1 = nbr + (long long)i1 * kD + c * 4;
  float* w2 = nbr + (long long)i2 * kD + c * 4;

  // nbr[f0] += v[f1]+v[f2]; nbr[f1] += v[f0]+v[f2]; nbr[f2] += v[f0]+v[f1];
  unsafeAtomicAdd(w0 + 0, a1.x + a2.x);
  unsafeAtomicAdd(w0 + 1, a1.y + a2.y);
  unsafeAtomicAdd(w0 + 2, a1.z + a2.z);
  unsafeAtomicAdd(w0 + 3, a1.w + a2.w);

  unsafeAtomicAdd(w1 + 0, a0.x + a2.x);
  unsafeAtomicAdd(w1 + 1, a0.y + a2.y);
  unsafeAtomicAdd(w1 + 2, a0.z + a2.z);
  unsafeAtomicAdd(w1 + 3, a0.w + a2.w);

  unsafeAtomicAdd(w2 + 0, a0.x + a1.x);
  unsafeAtomicAdd(w2 + 1, a0.y + a1.y);
  unsafeAtomicAdd(w2 + 2, a0.z + a1.z);
  unsafeAtomicAdd(w2 + 3, a0.w + a1.w);

  // Each face corner receives two directed edges -> degree += 2.
  if (c == 0) {
    unsafeAtomicAdd(deg + i0, 2.0f);
    unsafeAtomicAdd(deg + i1, 2.0f);
    unsafeAtomicAdd(deg + i2, 2.0f);
  }
}

// -------------------------------------------------------------------------
// Kernel 3: finalize in place: out = (deg*v - nbr) / (deg + 1e-12).
// Last-use streaming -> non-temporal loads/stores (TH=NT) so this pass
// does not evict the L2-resident gather/atomic working set.
// -------------------------------------------------------------------------
__global__ void ul_final_kernel(const float* __restrict__ verts,
                                const float* __restrict__ deg,
                                float*       __restrict__ out) {
  const long long idx4 = (long long)blockIdx.x * blockDim.x + threadIdx.x;
  if (idx4 >= kVertsTotal * kC4) return;

  const int   vtx = (int)(idx4 >> 4);
  const float dg  = deg[vtx];
  const float r   = 1.0f / (dg + 1e-12f);

  const f32x4* __restrict__ v4 = (const f32x4*)verts;
  f32x4*       __restrict__ o4 = (f32x4*)out;

  const f32x4 v  = __builtin_nontemporal_load(&v4[idx4]);
  const f32x4 nb = __builtin_nontemporal_load(&o4[idx4]);
  const f32x4 x  = (dg * v - nb) * r;
  __builtin_nontemporal_store(x, &o4[idx4]);
}

// -------------------------------------------------------------------------
extern "C" void kernel_launch(void* const* d_in, const int* in_sizes, int n_in,
                              void* d_out, int out_size, void* d_ws, size_t ws_size,
                              hipStream_t stream) {
  const float* verts = (const float*)d_in[0];   // [B, NV, D] f32
  const int*   faces = (const int*)d_in[1];     // [B, F, 3] i32
  float*       out   = (float*)d_out;           // [B, NV, D] f32; doubles as nbr accum
  float*       deg   = (float*)d_ws;            // [B*NV] f32 degree accum (1.6 MB)

  (void)in_sizes; (void)n_in; (void)out_size; (void)ws_size;

  // 1) zero accumulators (out reused as nbr; deg in workspace)
  {
    const int nOut4 = (int)(kOutFloats / 4);    // 6,400,000
    const int nDeg4 = (int)(kVertsTotal / 4);   // 100,000
    ul_zero_kernel<<<2048, kBlock, 0, stream>>>((float4*)out, nOut4,
                                                (float4*)deg, nDeg4);
  }

  // 2) scatter-accumulate neighbor sums + degrees (16 faces per block, no tail)
  {
    const int grid = (int)(kFacesTotal / kFacesPerBlk);  // 50,000
    ul_accum_kernel<<<grid, kBlock, 0, stream>>>(verts, faces, out, deg);
  }

  // 3) finalize: out = (deg*v - nbr) / (deg + eps)
  {
    const long long nThreads = kVertsTotal * kC4;        // 6,400,000
    const int grid = (int)((nThreads + kBlock - 1) / kBlock);
    ul_final_kernel<<<grid, kBlock, 0, stream>>>(verts, deg, out);
  }
}